// CondConv2d_57269093925141
// MI455X (gfx1250) — compile-verified
//
#include <hip/hip_runtime.h>
#include <math.h>

typedef __attribute__((ext_vector_type(2))) float v2f;
typedef __attribute__((ext_vector_type(8))) float v8f;

#define BATCH 16
#define CIN   128
#define COUT  128
#define KEXP  4
#define KS    3
#define HH    128
#define WW    128
#define HID   32

#define TILE_N 64
// LDS patch: 8 channel-pairs x 3 rows x 80 cols of float2.
// pair stride = 240 float2 = 480 dwords == 32 mod 64 banks -> the two
// lane-halves (half=0/1 read pair+0/pair+1) hit disjoint bank halves.
#define P_ROW  80
#define P_PAIR (3 * P_ROW)     // 240 float2 per channel-pair
#define P_ELEM (8 * 3 * 66)    // 1584 staged float2 per chunk
#define NST    7               // ceil(1584/256) staging regs per thread

// ---------------- Kernel 1: global average pool -> g[b*CIN+c] ----------------
__global__ void gap_kernel(const float* __restrict__ x, float* __restrict__ g) {
    __shared__ float red[256];
    const int bc = blockIdx.x;                      // b*CIN + c
    const float4* p = (const float4*)(x + (size_t)bc * (HH * WW));
    float s = 0.f;
    for (int i = threadIdx.x; i < (HH * WW) / 4; i += 256) {
        float4 v = p[i];
        s += v.x + v.y + v.z + v.w;
    }
    red[threadIdx.x] = s;
    __syncthreads();
    for (int off = 128; off > 0; off >>= 1) {
        if (threadIdx.x < off) red[threadIdx.x] += red[threadIdx.x + off];
        __syncthreads();
    }
    if (threadIdx.x == 0) g[bc] = red[0] * (1.0f / (HH * WW));
}

// ---------------- Kernel 2: router MLP + softmax + beff ----------------
__global__ void router_kernel(const float* __restrict__ g,
                              const float* __restrict__ w1, const float* __restrict__ b1,
                              const float* __restrict__ w2, const float* __restrict__ b2,
                              const float* __restrict__ bias,
                              float* __restrict__ alpha, float* __restrict__ beff) {
    const int b = blockIdx.x;
    const int j = threadIdx.x;          // 0..31 (one wave32)
    __shared__ float hs[HID];
    __shared__ float lg[KEXP];
    __shared__ float al[KEXP];

    float acc = b1[j];
    for (int c = 0; c < CIN; ++c) acc += g[b * CIN + c] * w1[j * CIN + c];
    hs[j] = fmaxf(acc, 0.f);
    __syncthreads();

    if (j < KEXP) {
        float l = b2[j];
        for (int t = 0; t < HID; ++t) l += hs[t] * w2[j * HID + t];
        lg[j] = l;
    }
    __syncthreads();

    if (j == 0) {
        float m = lg[0];
        for (int k = 1; k < KEXP; ++k) m = fmaxf(m, lg[k]);
        float s = 0.f;
        for (int k = 0; k < KEXP; ++k) { al[k] = __expf(lg[k] - m); s += al[k]; }
        float inv = 1.f / s;
        for (int k = 0; k < KEXP; ++k) { al[k] *= inv; alpha[b * KEXP + k] = al[k]; }
    }
    __syncthreads();

    for (int o = j; o < COUT; o += 32) {
        float s = 0.f;
        for (int k = 0; k < KEXP; ++k) s += al[k] * bias[k * COUT + o];
        beff[b * COUT + o] = s;
    }
}

// -------- Kernel 3: mix experts -> weff2[b][tap][ci/2][co] (float2 over ci%2) --------
__global__ void weff_kernel(const float* __restrict__ alpha,
                            const float* __restrict__ experts,
                            float* __restrict__ weff2) {
    int idx = blockIdx.x * 256 + threadIdx.x;   // flat float index over B*9*64*128*2
    int v   = idx & 1;                          // ci % 2
    int co  = (idx >> 1) & (COUT - 1);
    int cp  = (idx >> 8) & 63;                  // ci / 2
    int t   = idx >> 14;
    int tap = t % (KS * KS);
    int b   = t / (KS * KS);
    int ci  = cp * 2 + v;

    float a0 = alpha[b * KEXP + 0];
    float a1 = alpha[b * KEXP + 1];
    float a2 = alpha[b * KEXP + 2];
    float a3 = alpha[b * KEXP + 3];

    size_t e = ((size_t)co * CIN + ci) * (KS * KS) + tap;
    const size_t estr = (size_t)COUT * CIN * KS * KS;
    float s = a0 * experts[e] + a1 * experts[e + estr]
            + a2 * experts[e + 2 * estr] + a3 * experts[e + 3 * estr];
    weff2[idx] = s;
}

// ------- Kernel 4: implicit-GEMM conv, fp32 WMMA, double-buffered LDS pipeline -------
__global__ void __launch_bounds__(256)
condconv_kernel(const float* __restrict__ x, const float* __restrict__ weff2,
                const float* __restrict__ beff, float* __restrict__ out) {
    __shared__ v2f patch[2][8 * P_PAIR];        // 2 x 15360 B

    const int blk = blockIdx.x;
    const int tilesPerRow = WW / TILE_N;        // 2
    const int tilesPerImg = HH * tilesPerRow;   // 256
    const int b  = blk / tilesPerImg;
    const int r  = blk % tilesPerImg;
    const int y  = r / tilesPerRow;
    const int x0 = (r % tilesPerRow) * TILE_N;

    const int tid  = threadIdx.x;
    const int wave = tid >> 5;                  // 0..7 -> C_out block of 16
    const int lane = tid & 31;
    const int m    = lane & 15;
    const int half = lane >> 4;                 // 0 or 1

    // weff2 as float2: element index = ((b*9 + tap)*64 + cp)*128 + co
    const v2f* wA = (const v2f*)weff2
                  + (size_t)b * 9 * 64 * COUT
                  + (size_t)half * COUT        // cp += half
                  + (size_t)(wave * 16 + m);   // co (lane-coalesced)
    const float* x_b = x + (size_t)b * CIN * HH * WW;

    // ---- precompute staging addresses once (chunk-invariant) ----
    v2f      sv[NST];
    int      ldsoff[NST];                       // v2f units within a buffer
    int      goff[NST];                         // float units, without channel-chunk term
    unsigned okmask = 0;
    #pragma unroll
    for (int k = 0; k < NST; ++k) {
        int i    = tid + k * 256;
        int col  = i % 66;
        int row  = (i / 66) % 3;
        int pair = i / 198;
        int yy = y + row - 1;
        int xx = x0 + col - 1;
        bool ok = (i < P_ELEM) && ((unsigned)yy < (unsigned)HH)
                               && ((unsigned)xx < (unsigned)WW);
        if (ok) okmask |= (1u << k);
        ldsoff[k] = (pair * 3 + row) * P_ROW + col;
        goff[k]   = ((2 * pair) * HH + yy) * WW + xx;
    }

    auto stage_load = [&](int cb) {             // global -> regs (latency hidden)
        #pragma unroll
        for (int k = 0; k < NST; ++k) {
            v2f v = {0.f, 0.f};
            if (okmask & (1u << k)) {
                const float* s0 = x_b + (size_t)cb * (HH * WW) + goff[k];
                v[0] = s0[0];
                v[1] = s0[HH * WW];
            }
            sv[k] = v;
        }
    };
    auto stage_store = [&](v2f* buf) {          // regs -> LDS (after compute)
        #pragma unroll
        for (int k = 0; k < NST; ++k) {
            if (tid + k * 256 < P_ELEM) buf[ldsoff[k]] = sv[k];
        }
    };

    v8f acc[4] = {};                            // 4 N-subtiles of 16x16 fp32

    stage_load(0);
    stage_store(patch[0]);

    for (int cbi = 0; cbi < 8; ++cbi) {
        __syncthreads();                        // patch[cbi&1] ready for all waves
        if (cbi < 7) stage_load((cbi + 1) * 16);

        const v2f* pB  = patch[cbi & 1] + half * P_PAIR + m;
        const v2f* wAc = wA + (size_t)(cbi * 8) * COUT;   // cp base for this chunk

        #pragma unroll
        for (int tap = 0; tap < KS * KS; ++tap) {
            const int ky = tap / 3, kx = tap % 3;
            #pragma unroll
            for (int kc = 0; kc < 4; ++kc) {
                v2f a = wAc[(tap * 64 + kc * 2) * COUT];  // one b64, imm offset
                #pragma unroll
                for (int t = 0; t < 4; ++t) {
                    v2f bf = pB[kc * 2 * P_PAIR + ky * P_ROW + t * 16 + kx];
                    acc[t] = __builtin_amdgcn_wmma_f32_16x16x4_f32(
                        false, a, false, bf, (short)0, acc[t], false, false);
                }
            }
        }

        if (cbi < 7) stage_store(patch[(cbi + 1) & 1]);   // safe: buffer last read
    }                                                     // before this barrier

    // store out[b][co][y][x0 + t*16 + m] + beff  (non-temporal: keep x in L2)
    #pragma unroll
    for (int t = 0; t < 4; ++t) {
        #pragma unroll
        for (int rr = 0; rr < 8; ++rr) {
            int co = wave * 16 + rr + 8 * half;
            float v = acc[t][rr] + beff[b * COUT + co];
            __builtin_nontemporal_store(
                v, &out[(((size_t)b * COUT + co) * HH + y) * WW + x0 + t * 16 + m]);
        }
    }
}

// ---------------- launcher ----------------
extern "C" void kernel_launch(void* const* d_in, const int* in_sizes, int n_in,
                              void* d_out, int out_size, void* d_ws, size_t ws_size,
                              hipStream_t stream) {
    (void)in_sizes; (void)n_in; (void)out_size; (void)ws_size;
    const float* x       = (const float*)d_in[0];
    const float* experts = (const float*)d_in[1];
    const float* bias    = (const float*)d_in[2];
    const float* w1      = (const float*)d_in[3];
    const float* b1      = (const float*)d_in[4];
    const float* w2      = (const float*)d_in[5];
    const float* b2      = (const float*)d_in[6];
    float* out = (float*)d_out;

    float* ws    = (float*)d_ws;
    float* g     = ws;                 // 2048
    float* alpha = ws + 2048;          // 64
    float* beff  = ws + 2112;          // 2048
    float* weff  = ws + 4160;          // 16*9*64*128*2 = 2359296 floats

    gap_kernel<<<BATCH * CIN, 256, 0, stream>>>(x, g);
    router_kernel<<<BATCH, 32, 0, stream>>>(g, w1, b1, w2, b2, bias, alpha, beff);
    weff_kernel<<<(BATCH * KS * KS * CIN * COUT) / 256, 256, 0, stream>>>(alpha, experts, weff);
    condconv_kernel<<<BATCH * HH * (WW / TILE_N), 256, 0, stream>>>(x, weff, beff, out);
}